// GATEncoder_10823317586228
// MI455X (gfx1250) — compile-verified
//
#include <hip/hip_runtime.h>
#include <hip/hip_bf16.h>
#include <math.h>

typedef __attribute__((ext_vector_type(2))) float v2f;
typedef __attribute__((ext_vector_type(8))) float v8f;

#define NNODES 20000
#define NEDGES 320000
#define EPAD   (NNODES + NEDGES)   /* 340000 edges incl. self loops */
#define SLOPE  0.2f

// ---------------------------------------------------------------------------
// GEMM: H[M x NC] = X[M x K] * W[K x NC]  (M=20000, K=128, NC in {128,64})
// One wave32 per 16x16 output tile, f32 WMMA, K-steps of 4.
// A-frag (16x4 f32): VGPR0 = K={0 | 2} by lane-half, VGPR1 = K={1 | 3}.
// B-frag (4x16 f32): mirrored. C/D: VGPR j holds row j (lanes 0-15) / j+8.
// ---------------------------------------------------------------------------
__global__ void gemm_wmma_f32(const float* __restrict__ X,
                              const float* __restrict__ W,
                              float* __restrict__ H,
                              int K, int NC) {
  const int row0 = blockIdx.x * 16;
  const int col0 = blockIdx.y * 16;
  const int lane = threadIdx.x;        // 0..31, full wave => EXEC all ones
  const int half = lane >> 4;          // 0 or 1
  const int mr   = lane & 15;

  v8f acc = {};
  for (int k = 0; k < K; k += 4) {
    const int ka = k + 2 * half;
    v2f a, b;
    a.x = X[(row0 + mr) * K + ka + 0];
    a.y = X[(row0 + mr) * K + ka + 1];
    b.x = W[(ka + 0) * NC + col0 + mr];
    b.y = W[(ka + 1) * NC + col0 + mr];
    acc = __builtin_amdgcn_wmma_f32_16x16x4_f32(
        /*neg_a=*/false, a, /*neg_b=*/false, b,
        /*c_mod=*/(short)0, acc, /*reuse_a=*/false, /*reuse_b=*/false);
  }
#pragma unroll
  for (int j = 0; j < 8; ++j) {
    H[(row0 + j + 8 * half) * NC + col0 + mr] = acc[j];
  }
}

// ---------------------------------------------------------------------------
__global__ void fill_zero(float* __restrict__ p, int n) {
  int t = blockIdx.x * blockDim.x + threadIdx.x;
  for (; t < n; t += gridDim.x * blockDim.x) p[t] = 0.0f;
}

__global__ void recip_inplace(float* __restrict__ p, int n) {
  int t = blockIdx.x * blockDim.x + threadIdx.x;
  if (t < n) p[t] = 1.0f / p[t];
}

// al_s[n,h] = sum_c H[n,h,c]*a_src[h,c];   al_d likewise.
__global__ void node_scores(const float* __restrict__ Hm,
                            const float* __restrict__ a_src,
                            const float* __restrict__ a_dst,
                            float* __restrict__ als, float* __restrict__ ald,
                            int heads, int C) {
  int t = blockIdx.x * blockDim.x + threadIdx.x;
  if (t >= NNODES * heads) return;
  const int node = t / heads;
  const int hd   = t % heads;
  const float* hp = Hm + (node * heads + hd) * C;
  const float* as = a_src + hd * C;
  const float* ad = a_dst + hd * C;
  float s = 0.0f, d = 0.0f;
  for (int c = 0; c < C; ++c) {
    float v = hp[c];
    s += v * as[c];
    d += v * ad[c];
  }
  als[t] = s;
  ald[t] = d;
}

__device__ __forceinline__ void edge_nodes(const int* __restrict__ ei, int e,
                                           int& s, int& d) {
  if (e < NEDGES) { s = ei[e]; d = ei[NEDGES + e]; }
  else            { s = e - NEDGES; d = s; }          // self loops
}

// exp(leaky_relu(al_s[src]+al_d[dst])) per (edge, head); accumulate denominator.
__global__ void edge_exp(const float* __restrict__ als,
                         const float* __restrict__ ald,
                         const int* __restrict__ ei,
                         float* __restrict__ expe, float* __restrict__ denom,
                         int heads) {
  int t = blockIdx.x * blockDim.x + threadIdx.x;
  if (t >= EPAD * heads) return;
  const int e  = t / heads;
  const int hd = t % heads;
  int s, d; edge_nodes(ei, e, s, d);
  float v = als[s * heads + hd] + ald[d * heads + hd];
  v = (v > 0.0f) ? v : SLOPE * v;        // leaky relu
  float p = expf(v);                     // softmax is shift-invariant
  expe[t] = p;
  atomicAdd(&denom[d * heads + hd], p);
}

// expe[t] -> alpha[t] = expe[t] * invden[dst, head]
__global__ void edge_scale(float* __restrict__ expe,
                           const float* __restrict__ invden,
                           const int* __restrict__ ei, int heads) {
  int t = blockIdx.x * blockDim.x + threadIdx.x;
  if (t >= EPAD * heads) return;
  const int e  = t / heads;
  const int hd = t % heads;
  int s, d; edge_nodes(ei, e, s, d);
  (void)s;
  expe[t] *= invden[d * heads + hd];
}

// ---------------------------------------------------------------------------
// One wave per edge; each lane owns a contiguous VEC-chunk of the OC=32*VEC
// channels (vectorized b128/b64 gather of H[src]); head index is constant per
// lane, so alpha is a single load; hot loop = pure v_fma + global_atomic_add.
// ---------------------------------------------------------------------------
template <int VEC> struct VecT;
template <> struct VecT<4> { using T = float4; };
template <> struct VecT<2> { using T = float2; };

template <int VEC>
__global__ void edge_aggregate(const float* __restrict__ Hm,
                               const float* __restrict__ alpha,
                               const int* __restrict__ ei,
                               float* __restrict__ out,
                               int heads, int cshift) {
  const int edge = blockIdx.x * (blockDim.x >> 5) + (threadIdx.x >> 5);
  const int lane = threadIdx.x & 31;
  if (edge >= EPAD) return;
  int s, d; edge_nodes(ei, edge, s, d);
  const int OC = 32 * VEC;
  const int colbase = lane * VEC;
  const float a = alpha[edge * heads + (colbase >> cshift)];
  using T = typename VecT<VEC>::T;
  const T hv = *(const T*)(Hm + s * OC + colbase);
  float* op = out + d * OC + colbase;
  const float* hp = (const float*)&hv;
#pragma unroll
  for (int v = 0; v < VEC; ++v) atomicAdd(op + v, hp[v] * a);
}

// y = acc + bias, optional ELU (last layer: heads==1 so head-mean is identity)
__global__ void bias_act(const float* __restrict__ acc,
                         const float* __restrict__ b,
                         float* __restrict__ y, int n, int NC, int do_elu) {
  int t = blockIdx.x * blockDim.x + threadIdx.x;
  if (t >= n) return;
  float v = acc[t] + b[t % NC];
  if (do_elu) v = (v > 0.0f) ? v : (expf(v) - 1.0f);
  y[t] = v;
}

// ---------------------------------------------------------------------------
extern "C" void kernel_launch(void* const* d_in, const int* in_sizes, int n_in,
                              void* d_out, int out_size, void* d_ws, size_t ws_size,
                              hipStream_t stream) {
  (void)in_sizes; (void)n_in; (void)out_size; (void)ws_size;
  const float* x  = (const float*)d_in[0];
  const int*   ei = (const int*)d_in[1];

  float* ws   = (float*)d_ws;
  float* xb0  = ws;                          // N*128
  float* xb1  = xb0  + NNODES * 128;         // N*128
  float* hbuf = xb1  + NNODES * 128;         // N*128
  float* accb = hbuf + NNODES * 128;         // N*128
  float* als  = accb + NNODES * 128;         // N*4
  float* ald  = als  + NNODES * 4;           // N*4
  float* den  = ald  + NNODES * 4;           // N*4
  float* expe = den  + NNODES * 4;           // EPAD*4

  const float* xin = x;
  for (int layer = 0; layer < 5; ++layer) {
    const float* W  = (const float*)d_in[2 + 4 * layer + 0];
    const float* as = (const float*)d_in[2 + 4 * layer + 1];
    const float* ad = (const float*)d_in[2 + 4 * layer + 2];
    const float* b  = (const float*)d_in[2 + 4 * layer + 3];
    const int heads  = (layer == 4) ? 1 : 4;
    const int C      = (layer == 4) ? 64 : 32;
    const int cshift = (layer == 4) ? 6 : 5;
    const int NC     = heads * C;
    const int K      = 128;                   // fan-in is 128 for every layer

    dim3 gg(NNODES / 16, NC / 16);
    gemm_wmma_f32<<<gg, 32, 0, stream>>>(xin, W, hbuf, K, NC);

    const int nacc = NNODES * NC;
    const int nden = NNODES * heads;
    const int nedg = EPAD * heads;
    fill_zero<<<(nacc + 255) / 256, 256, 0, stream>>>(accb, nacc);
    fill_zero<<<(nden + 255) / 256, 256, 0, stream>>>(den, nden);

    node_scores<<<(nden + 127) / 128, 128, 0, stream>>>(
        hbuf, as, ad, als, ald, heads, C);

    edge_exp<<<(nedg + 255) / 256, 256, 0, stream>>>(
        als, ald, ei, expe, den, heads);

    recip_inplace<<<(nden + 255) / 256, 256, 0, stream>>>(den, nden);

    edge_scale<<<(nedg + 255) / 256, 256, 0, stream>>>(expe, den, ei, heads);

    if (NC == 128)
      edge_aggregate<4><<<(EPAD + 7) / 8, 256, 0, stream>>>(
          hbuf, expe, ei, accb, heads, cshift);
    else
      edge_aggregate<2><<<(EPAD + 7) / 8, 256, 0, stream>>>(
          hbuf, expe, ei, accb, heads, cshift);

    float* yout = (layer == 4) ? (float*)d_out : ((layer & 1) ? xb1 : xb0);
    bias_act<<<(nacc + 255) / 256, 256, 0, stream>>>(
        accb, b, yout, nacc, NC, (layer < 4) ? 1 : 0);
    xin = yout;
  }
}